// InputLayer_7017976562084
// MI455X (gfx1250) — compile-verified
//
#include <hip/hip_runtime.h>

// Belief-propagation decoder input layer:
//   out[b,n] = log((1+z+eps)/(1-z+eps)),
//   z[b,n]   = prod_i ( tanh(0.5*clip(x[b,i]*W[i,n], +-10)) + exp(-1e6*|u|) )
// Product reduction over i (no contraction -> no WMMA). Compute-bound on
// transcendentals; gfx1250 async-to-LDS staging + v_tanh_f32/v_exp_f32.

#define N_BITS   128
#define NEURONS  512
#define NTILE    256     // columns per block (== blockDim.x)
#define ROWS     8       // batch rows per block
#define EPS      1e-8f

#define AS1 __attribute__((address_space(1)))
#define AS3 __attribute__((address_space(3)))

typedef int v4i __attribute__((vector_size(16)));

__device__ __forceinline__ float edge_term(float p) {
  // p = 0.5 * x * w  (x pre-scaled by 0.5 at staging), clamp +-5 == clip(xw,+-10)/2
  float c = fminf(fmaxf(p, -5.0f), 5.0f);
#if __has_builtin(__builtin_amdgcn_tanhf)
  float u = __builtin_amdgcn_tanhf(c);             // gfx1250 v_tanh_f32, 1 trans op
#else
  float e = __expf(2.0f * c);                      // tanh(c) = (e^{2c}-1)/(e^{2c}+1)
  float u = (e - 1.0f) * __builtin_amdgcn_rcpf(e + 1.0f);
#endif
  float g = __expf(-1.0e6f * fabsf(u));            // v_exp_f32, underflows to 0
  return u + g;
}

__global__ __launch_bounds__(NTILE)
void bp_input_layer_kernel(const float* __restrict__ x,
                           const float* __restrict__ w,
                           float* __restrict__ out,
                           int batch) {
  __shared__ float lds_w[N_BITS * NTILE];   // 128 KB W tile, [i][n] layout
  __shared__ float lds_x[ROWS * N_BITS];    // 4 KB, pre-scaled by 0.5

  const int tid   = threadIdx.x;            // 0..255
  const int b0    = blockIdx.x * ROWS;
  const int nbase = blockIdx.y * NTILE;

  // ---- stage W tile into LDS: 32 straight-line async Global->LDS issues ----
  // j = tid + t*NTILE ; i = j>>6 = (tid>>6)+4t ; c = j&63 = tid&63
  {
    const float* gsrc = w + (tid >> 6) * NEURONS + nbase + 4 * (tid & 63);
    float*       ldst = &lds_w[4 * tid];
#if __has_builtin(__builtin_amdgcn_global_load_async_to_lds_b128)
    #pragma unroll
    for (int t = 0; t < (N_BITS * NTILE / 4) / NTILE; ++t) {   // 32, literal
      __builtin_amdgcn_global_load_async_to_lds_b128(
          (AS1 v4i*)(gsrc + t * 4 * NEURONS),    // +8 KB per step (4 W rows)
          (AS3 v4i*)(ldst + t * 4 * NTILE),      // +4 KB per step
          /*offset=*/0, /*cpol=*/0);
    }
#else
    #pragma unroll
    for (int t = 0; t < (N_BITS * NTILE / 4) / NTILE; ++t) {
      *(float4*)(ldst + t * 4 * NTILE) = *(const float4*)(gsrc + t * 4 * NEURONS);
    }
#endif
  }

  // ---- stage x rows (scaled by 0.5) through VGPRs: 4 fixed iterations ----
  #pragma unroll
  for (int t = 0; t < (ROWS * N_BITS) / NTILE; ++t) {          // 4, literal
    const int k = tid + t * NTILE;
    const int r = k >> 7;                   // k / N_BITS
    const int i = k & (N_BITS - 1);
    const int b = b0 + r;
    lds_x[k] = (b < batch) ? 0.5f * x[b * N_BITS + i] : 0.0f;
  }

#if __has_builtin(__builtin_amdgcn_s_wait_asynccnt)
  __builtin_amdgcn_s_wait_asynccnt(0);      // this wave's async loads done
#else
  asm volatile("s_wait_asynccnt 0" ::: "memory");
#endif
  __syncthreads();                          // cross-wave visibility

  // ---- product reduction: lane = one output column, loop over rows ----
  const int n    = tid;
  const int rem  = batch - b0;
  const int rows = rem < ROWS ? rem : ROWS;
  for (int r = 0; r < rows; ++r) {
    float a0 = 1.0f, a1 = 1.0f, a2 = 1.0f, a3 = 1.0f;  // 4 independent chains
    const float* xr = &lds_x[r * N_BITS];
    #pragma unroll 8
    for (int i = 0; i < N_BITS; i += 4) {
      const float4 xs = *(const float4*)(xr + i);      // wave-uniform broadcast
      a0 *= edge_term(xs.x * lds_w[(i + 0) * NTILE + n]);  // conflict-free ds reads
      a1 *= edge_term(xs.y * lds_w[(i + 1) * NTILE + n]);
      a2 *= edge_term(xs.z * lds_w[(i + 2) * NTILE + n]);
      a3 *= edge_term(xs.w * lds_w[(i + 3) * NTILE + n]);
    }
    const float z   = (a0 * a1) * (a2 * a3);
    const float num = 1.0f + z + EPS;
    const float den = 1.0f - z + EPS;
    out[(b0 + r) * NEURONS + nbase + n] =
        __logf(num * __builtin_amdgcn_rcpf(den));      // v_log_f32 + v_rcp_f32
  }
}

extern "C" void kernel_launch(void* const* d_in, const int* in_sizes, int n_in,
                              void* d_out, int out_size, void* d_ws, size_t ws_size,
                              hipStream_t stream) {
  const float* x = (const float*)d_in[0];        // [batch, 128] f32
  const float* w = (const float*)d_in[1];        // [128, 512] f32
  float* out = (float*)d_out;                    // [batch, 512] f32
  const int batch = in_sizes[0] / N_BITS;
  dim3 grid((batch + ROWS - 1) / ROWS, NEURONS / NTILE);
  bp_input_layer_kernel<<<grid, NTILE, 0, stream>>>(x, w, out, batch);
}